// MoEGate_79061757984863
// MI455X (gfx1250) — compile-verified
//
#include <hip/hip_runtime.h>
#include <hip/hip_bf16.h>

// MoE gate for MI455X (gfx1250): router GEMM via V_WMMA_F32_16X16X4_F32,
// fused softmax / top-2 / seq-aux loss.
//
// Shapes: x[T=16384, H=2048] f32, W[E=64, H=2048] f32.
// Out (flat f32): topk_idx[T,2] (as float), topk_weight[T,2], aux_loss (1).
//
// Geometry: 256 blocks x 128 threads (4 waves). Each wave owns a 16-token
// tile and all 64 experts (4 WMMA accumulator tiles). Weights are staged
// into LDS in K=64 chunks with register double-buffering so global loads
// overlap the WMMA loop; one barrier per chunk.

typedef __attribute__((ext_vector_type(2))) float v2f;
typedef __attribute__((ext_vector_type(8))) float v8f;

#define HDIM     2048
#define NEXP     64
#define TOPK     2
#define T_TOK    16384
#define S_LEN    4096
#define NBATCH   4
#define KCHUNK   64
#define NCHUNK   (HDIM / KCHUNK)       // 32
#define WSTRIDE  68                    // padded LDS stride (dwords): banks 4e apart
#define TOKPB    64                    // tokens per block
#define ALPHA    0.01f

__global__ __launch_bounds__(128) void moe_gate_main(
    const float* __restrict__ x,      // [T, H]
    const float* __restrict__ wgt,    // [E, H]
    float* __restrict__ out,          // [4*T + 1]
    float* __restrict__ g_ce,         // [B, E] ws (zeroed)
    float* __restrict__ g_sum)        // [B, E] ws (zeroed)
{
    // Double-buffered weight chunk: 2 x 64 x 68 dwords = 34816 B.
    // Reused as logits/probs [64 tokens x 64 experts] = 16384 B after GEMM.
    __shared__ float smem[2 * NEXP * WSTRIDE];
    __shared__ float lds_ce[NEXP];

    const int tid  = threadIdx.x;
    const int lane = tid & 31;
    const int wv   = tid >> 5;        // wave id 0..3
    const int hl   = lane >> 4;       // half-wave select (K pair / M+8)
    const int l16  = lane & 15;

    const long blk_tok = (long)blockIdx.x * TOKPB;
    const int  wv_tok  = wv * 16;
    const float* xrow  = x + (blk_tok + wv_tok + l16) * HDIM + 2 * hl;

    if (tid < NEXP) lds_ce[tid] = 0.0f;

    // Staging: chunk = 64 experts x 64 floats = 4096 floats; 128 threads
    // -> 32 floats = 8 float4 per thread. Thread i covers expert e = i>>4,
    // float4 index (i&15) within the row (rows fully coalesced, 16 thr/row).
    const int st_e  = tid >> 4;              // expert rows tid.., step 8
    const int st_k4 = (tid & 15) << 2;       // float offset 0..60
    const float* st_src = wgt + (long)st_e * HDIM + st_k4;

    v8f acc0 = {}, acc1 = {}, acc2 = {}, acc3 = {};
    float4 stg[8];

    // ---- prologue: stage chunk 0
#pragma unroll
    for (int j = 0; j < 8; ++j)
        stg[j] = *(const float4*)(st_src + (long)(j * 8) * HDIM);
#pragma unroll
    for (int j = 0; j < 8; ++j) {
        float* dst = &smem[(st_e + j * 8) * WSTRIDE + st_k4];
        dst[0] = stg[j].x; dst[1] = stg[j].y; dst[2] = stg[j].z; dst[3] = stg[j].w;
    }
    __syncthreads();

    for (int c = 0; c < NCHUNK; ++c) {
        const float* buf = &smem[(c & 1) * NEXP * WSTRIDE];
        const int kc = c * KCHUNK;

        // issue next chunk's global loads first (they fly behind the WMMAs)
        const bool more = (c + 1) < NCHUNK;
        if (more) {
#pragma unroll
            for (int j = 0; j < 8; ++j)
                stg[j] = *(const float4*)(st_src + (long)(j * 8) * HDIM + kc + KCHUNK);
        }

        // ---- WMMA K-loop: D(16x16) += A(16x4) x B(4x16), 4 expert tiles
#pragma unroll 4
        for (int kk = 0; kk < KCHUNK; kk += 4) {
            const int kcol = kk + 2 * hl;
            const v2f a  = *(const v2f*)(xrow + kc + kk);
            const v2f b0 = *(const v2f*)&buf[( 0 + l16) * WSTRIDE + kcol];
            const v2f b1 = *(const v2f*)&buf[(16 + l16) * WSTRIDE + kcol];
            const v2f b2 = *(const v2f*)&buf[(32 + l16) * WSTRIDE + kcol];
            const v2f b3 = *(const v2f*)&buf[(48 + l16) * WSTRIDE + kcol];
            acc0 = __builtin_amdgcn_wmma_f32_16x16x4_f32(false, a, false, b0, (short)0, acc0, false, false);
            acc1 = __builtin_amdgcn_wmma_f32_16x16x4_f32(false, a, false, b1, (short)0, acc1, false, false);
            acc2 = __builtin_amdgcn_wmma_f32_16x16x4_f32(false, a, false, b2, (short)0, acc2, false, false);
            acc3 = __builtin_amdgcn_wmma_f32_16x16x4_f32(false, a, false, b3, (short)0, acc3, false, false);
        }

        // write staged chunk into the other buffer, single barrier per chunk
        if (more) {
            float* nbuf = &smem[((c + 1) & 1) * NEXP * WSTRIDE];
#pragma unroll
            for (int j = 0; j < 8; ++j) {
                float* dst = &nbuf[(st_e + j * 8) * WSTRIDE + st_k4];
                dst[0] = stg[j].x; dst[1] = stg[j].y; dst[2] = stg[j].z; dst[3] = stg[j].w;
            }
        }
        __syncthreads();
    }

    // ---- spill logits to LDS: C/D layout -> [token_local][expert]
    float* lg = smem;  // [64][64]
#pragma unroll
    for (int r = 0; r < 8; ++r) {
        const int tl = wv_tok + r + 8 * hl;
        lg[tl * NEXP +  0 + l16] = acc0[r];
        lg[tl * NEXP + 16 + l16] = acc1[r];
        lg[tl * NEXP + 32 + l16] = acc2[r];
        lg[tl * NEXP + 48 + l16] = acc3[r];
    }
    __syncthreads();

    // ---- per-token softmax + top-2 (threads 0..63, one token each)
    if (tid < TOKPB) {
        float* row = &lg[tid * NEXP];
        float m1 = -3.4e38f, m2 = -3.4e38f;
        int   i1 = 0, i2 = 0;
#pragma unroll 8
        for (int e = 0; e < NEXP; ++e) {
            const float v = row[e];
            if (v > m1)      { m2 = m1; i2 = i1; m1 = v; i1 = e; }
            else if (v > m2) { m2 = v; i2 = e; }
        }
        float denom = 0.0f;
#pragma unroll 8
        for (int e = 0; e < NEXP; ++e) {
            const float p = __expf(row[e] - m1);
            row[e] = p;
            denom += p;
        }
        const float inv = 1.0f / denom;
#pragma unroll 8
        for (int e = 0; e < NEXP; ++e) row[e] *= inv;   // probs in place (for aux)

        const float tw0 = inv;                   // exp(0)*inv
        const float tw1 = __expf(m2 - m1) * inv;
        const float rs  = 1.0f / (tw0 + tw1 + 1e-20f);

        const long t = blk_tok + tid;
        out[2 * t]                 = (float)i1;
        out[2 * t + 1]             = (float)i2;
        out[2 * T_TOK + 2 * t]     = tw0 * rs;
        out[2 * T_TOK + 2 * t + 1] = tw1 * rs;

        atomicAdd(&lds_ce[i1], 1.0f);
        atomicAdd(&lds_ce[i2], 1.0f);
    }
    __syncthreads();

    // ---- block reduction of probs per expert + flush to global (b = block/64)
    if (tid < NEXP) {
        float s = 0.0f;
        for (int t = 0; t < TOKPB; ++t) s += lg[t * NEXP + tid];
        const int b = blockIdx.x >> 6;
        atomicAdd(&g_sum[b * NEXP + tid], s);
        atomicAdd(&g_ce [b * NEXP + tid], lds_ce[tid]);
    }
}

// aux = ALPHA * mean_b( sum_e( count[b,e] * E/(S*K) * mean_s(scores[b,s,e]) ) )
__global__ __launch_bounds__(256) void moe_gate_aux(
    const float* __restrict__ g_ce,
    const float* __restrict__ g_sum,
    float* __restrict__ out_aux)
{
    __shared__ float red[256];
    const int tid = threadIdx.x;                    // 256 = B*E
    const float ce_scaled = g_ce[tid] * ((float)NEXP / ((float)S_LEN * (float)TOPK));
    const float mean_s    = g_sum[tid] * (1.0f / (float)S_LEN);
    red[tid] = ce_scaled * mean_s;
    __syncthreads();
    for (int s = 128; s > 0; s >>= 1) {
        if (tid < s) red[tid] += red[tid + s];
        __syncthreads();
    }
    if (tid == 0) *out_aux = red[0] * (ALPHA / (float)NBATCH);
}

extern "C" void kernel_launch(void* const* d_in, const int* in_sizes, int n_in,
                              void* d_out, int out_size, void* d_ws, size_t ws_size,
                              hipStream_t stream) {
    const float* x   = (const float*)d_in[0];   // [4,4096,2048] f32
    const float* wgt = (const float*)d_in[1];   // [64,2048] f32
    float* out   = (float*)d_out;               // 4*T + 1 floats
    float* g_ce  = (float*)d_ws;                // [4,64]
    float* g_sum = g_ce + NBATCH * NEXP;        // [4,64]

    hipMemsetAsync(d_ws, 0, 2 * NBATCH * NEXP * sizeof(float), stream);
    moe_gate_main<<<T_TOK / TOKPB, 128, 0, stream>>>(x, wgt, out, g_ce, g_sum);
    moe_gate_aux<<<1, 256, 0, stream>>>(g_ce, g_sum, out + 4 * T_TOK);
}